// summ_18451179503737
// MI455X (gfx1250) — compile-verified
//
#include <hip/hip_runtime.h>
#include <stdint.h>

// Problem geometry (fixed by the reference: a = (8192, 4096) fp32, scan over dim0)
#define ROWS    8192
#define COLS    4096
#define COLS4   (COLS / 4)        // 1024 float4 columns
#define BLOCK   256               // 8 waves (wave32)
#define CGRPS   (COLS4 / BLOCK)   // 4 column-group blocks cover all columns
#define CHUNK   64                // rows per chunk
#define NCHUNK  (ROWS / CHUNK)    // 128 chunks -> 512 blocks * 8 waves = 4096 waves
#define STAGE   8                 // rows per async-LDS batch
#define NBATCH  (CHUNK / STAGE)   // 8 pipelined batches per chunk
#define P2BATCH 16                // chunk partials batched per scan step in pass 2

typedef float v4f __attribute__((ext_vector_type(4)));

// Types matching __builtin_amdgcn_global_load_async_to_lds_b128's signature:
//   param0: int __attribute__((vector_size(16))) __device__ * (global AS)
//   param1: matching shared-AS pointer
typedef int v4i_t __attribute__((vector_size(16)));
typedef __attribute__((address_space(1))) v4i_t* g_v4i_ptr;
typedef __attribute__((address_space(3))) v4i_t* l_v4i_ptr;

// ---- CDNA5 async global->LDS copy (ASYNCcnt-tracked), builtin or asm ----
__device__ __forceinline__ void async_ld_b128(const void* gptr, void* lds_generic) {
  // low 32 bits of a generic LDS pointer == wave-relative LDS byte offset
  uint32_t loff = (uint32_t)(uintptr_t)lds_generic;
#if __has_builtin(__builtin_amdgcn_global_load_async_to_lds_b128)
  __builtin_amdgcn_global_load_async_to_lds_b128(
      (g_v4i_ptr)(void*)gptr, (l_v4i_ptr)loff, /*offset=*/0, /*cpol=*/0);
#else
  uint64_t ga = (uint64_t)(uintptr_t)gptr;
  asm volatile("global_load_async_to_lds_b128 %0, %1, off"
               :: "v"(loff), "v"(ga)
               : "memory");
#endif
}

template <int N>
__device__ __forceinline__ void wait_async_le() {
#if __has_builtin(__builtin_amdgcn_s_wait_asynccnt)
  __builtin_amdgcn_s_wait_asynccnt(N);
#else
  asm volatile("s_wait_asynccnt %0" :: "i"(N) : "memory");
#endif
}

__device__ __forceinline__ void wait_ds0() {
  asm volatile("s_wait_dscnt 0x0" ::: "memory");  // prior LDS reads retired
}

// ---- Pass 1: per-(chunk, column) partial sums.
// Rows streamed HBM -> LDS with double-buffered async copies (ASYNCcnt pipeline),
// wave-private staging slabs so no cross-wave barriers are needed.
__global__ void __launch_bounds__(BLOCK)
scan_pass1(const float* __restrict__ a, float* __restrict__ part) {
  __shared__ v4f stage[2][BLOCK / 32][STAGE][32];   // 64 KB, double-buffered
  const int lane = threadIdx.x & 31;
  const int wv   = threadIdx.x >> 5;
  const int c4   = blockIdx.x * BLOCK + threadIdx.x;  // float4 column index
  const int row0 = blockIdx.y * CHUNK;

  const v4f* __restrict__ a4 = (const v4f*)a;
  v4f acc = (v4f)0.0f;

  // prologue: batch 0 -> buffer 0
#pragma unroll
  for (int s = 0; s < STAGE; ++s)
    async_ld_b128(&a4[(size_t)(row0 + s) * COLS4 + c4], &stage[0][wv][s][lane]);

  for (int k = 0; k < NBATCH; ++k) {
    const int cur = k & 1;
    if (k + 1 < NBATCH) {
      wait_ds0();  // buffer (k+1)&1 fully consumed before async engine rewrites it
#pragma unroll
      for (int s = 0; s < STAGE; ++s)
        async_ld_b128(&a4[(size_t)(row0 + (k + 1) * STAGE + s) * COLS4 + c4],
                      &stage[cur ^ 1][wv][s][lane]);
      wait_async_le<STAGE>();   // in-order: batch k complete, batch k+1 in flight
    } else {
      wait_async_le<0>();
    }
#pragma unroll
    for (int s = 0; s < STAGE; ++s)
      acc += stage[cur][wv][s][lane];
  }
  ((v4f*)part)[(size_t)blockIdx.y * COLS4 + c4] = acc;   // RT store: stays in L2
}

// ---- Pass 2: in-place exclusive scan of the 128 chunk partials per column.
// Batched loads (16 independent per step) keep the dependent chain at 8 memory
// round-trips instead of 128 (only 32 waves run here, so latency matters).
__global__ void __launch_bounds__(BLOCK)
scan_pass2(float* __restrict__ part) {
  const int c4 = blockIdx.x * BLOCK + threadIdx.x;  // 0..COLS4-1
  v4f* p4 = (v4f*)part;
  v4f run = (v4f)0.0f;
  v4f vals[P2BATCH];

  for (int b = 0; b < NCHUNK; b += P2BATCH) {
#pragma unroll
    for (int i = 0; i < P2BATCH; ++i)
      vals[i] = p4[(size_t)(b + i) * COLS4 + c4];
#pragma unroll
    for (int i = 0; i < P2BATCH; ++i) {
      v4f v = vals[i];
      p4[(size_t)(b + i) * COLS4 + c4] = run;   // exclusive
      run += v;
    }
  }
}

// ---- Pass 3: re-read input (L2-resident after pass 1), emit exclusive scan.
// Output is streamed with non-temporal stores so the 128 MiB result does not
// evict the input from the 192 MB L2.
__global__ void __launch_bounds__(BLOCK)
scan_pass3(const float* __restrict__ a, const float* __restrict__ part,
           float* __restrict__ out) {
  const int c4   = blockIdx.x * BLOCK + threadIdx.x;
  const int row0 = blockIdx.y * CHUNK;
  const v4f* __restrict__ a4 = (const v4f*)a;
  const v4f* __restrict__ p4 = (const v4f*)part;
  v4f* __restrict__ o4 = (v4f*)out;

  v4f run = p4[(size_t)blockIdx.y * COLS4 + c4];

#pragma unroll 4
  for (int r = 0; r < CHUNK; ++r) {
    size_t idx = (size_t)(row0 + r) * COLS4 + c4;
    // prefetch a few rows ahead (clamped); emits global_prefetch_b8
    int pr = row0 + r + 8;
    pr = pr < ROWS ? pr : ROWS - 1;
    __builtin_prefetch(&a4[(size_t)pr * COLS4 + c4], 0, 1);

    v4f v = a4[idx];
    __builtin_nontemporal_store(run, &o4[idx]);
    run += v;
  }
}

extern "C" void kernel_launch(void* const* d_in, const int* in_sizes, int n_in,
                              void* d_out, int out_size, void* d_ws, size_t ws_size,
                              hipStream_t stream) {
  const float* a = (const float*)d_in[0];
  float* out  = (float*)d_out;
  float* part = (float*)d_ws;   // NCHUNK * COLS * 4B = 2 MiB scratch

  dim3 blk(BLOCK);
  scan_pass1<<<dim3(CGRPS, NCHUNK), blk, 0, stream>>>(a, part);
  scan_pass2<<<dim3(CGRPS), blk, 0, stream>>>(part);
  scan_pass3<<<dim3(CGRPS, NCHUNK), blk, 0, stream>>>(a, part, out);
}